// FeedForwardQuantum_65481071397245
// MI455X (gfx1250) — compile-verified
//
#include <hip/hip_runtime.h>
#include <math.h>

typedef float v2f __attribute__((ext_vector_type(2)));
typedef float v8f __attribute__((ext_vector_type(8)));

#define NQ     10
#define QDIM   1024          // 2^10 amplitudes
#define EMBED  256
#define FFN    1024
#define HPAD   (FFN + 4)     // LDS row pad -> conflict-free A-frag reads
#define ROWS   32            // token rows per FFN workgroup (M-tile)

// ---------------------------------------------------------------------------
// Kernel 1: 10-qubit state-vector sim, one 256-thread block per token.
//   * data-RX + layer-0 RX folded: product state built analytically
//   * CNOT ring #1 folded into the initial scatter (prefix-XOR permutation)
//   * 10 in-place RX passes (layer-1 params) on LDS state
//   * CNOT ring #2 folded into the measurement signs
// ---------------------------------------------------------------------------
__device__ __forceinline__ unsigned ring_perm(unsigned j) {
    // wire w lives at bit (9-w); ring CNOT(w,(w+1)%10) => prefix-XOR over wires
    unsigned q = j;
    q ^= q >> 1; q ^= q >> 2; q ^= q >> 4; q ^= q >> 8;
    // bits 0..8 (wires 9..1) correct; wire 0 (bit 9) = XOR(wires 1..9)
    return (q & 0x1FFu) | ((((q & 1u) ^ (j >> 9)) & 1u) << 9);
}

__global__ __launch_bounds__(256)
void qsim_kernel(const float* __restrict__ x, const float* __restrict__ qp,
                 float* __restrict__ meas)
{
    __shared__ float sr[QDIM], si[QDIM];
    __shared__ float c1[NQ], s1[NQ], c2[NQ], s2[NQ];
    __shared__ float red[8][NQ];

    const int t   = blockIdx.x;
    const int tid = threadIdx.x;

    if (tid < NQ) {
        const float phi = x[(size_t)t * EMBED + tid] + qp[tid]; // fold layer-0 RX
        c1[tid] = cosf(0.5f * phi);  s1[tid] = sinf(0.5f * phi);
        c2[tid] = cosf(0.5f * qp[NQ + tid]);
        s2[tid] = sinf(0.5f * qp[NQ + tid]);
    }
    __syncthreads();

    // ---- analytic product state, scattered through ring-permutation #1 ----
    for (int j = tid; j < QDIM; j += 256) {
        float mag = (j & 512) ? s1[0] : c1[0];
        #pragma unroll
        for (int w = 1; w < NQ; ++w)
            mag *= ((j >> (9 - w)) & 1) ? s1[w] : c1[w];
        const int pc = __popc((unsigned)j) & 3;   // amp = (-i)^pc * mag
        const float re = (pc == 0) ? mag : ((pc == 2) ? -mag : 0.0f);
        const float im = (pc == 1) ? -mag : ((pc == 3) ? mag : 0.0f);
        const unsigned p = ring_perm((unsigned)j);
        sr[p] = re;  si[p] = im;
    }
    __syncthreads();

    // ---- layer-1 RX gates, in place (pairs disjoint per gate) ----
    #pragma unroll
    for (int w = 0; w < NQ; ++w) {
        const int m = 1 << (9 - w);
        const float c = c2[w], s = s2[w];
        #pragma unroll
        for (int pp = 0; pp < 2; ++pp) {
            const int p  = tid + pp * 256;            // 512 pairs
            const int i0 = ((p & ~(m - 1)) << 1) | (p & (m - 1));
            const int i1 = i0 + m;
            const float ar = sr[i0], ai = si[i0];
            const float br = sr[i1], bi = si[i1];
            // a' = c*a - i*s*b ; b' = -i*s*a + c*b
            sr[i0] = fmaf(c, ar,  s * bi);
            si[i0] = fmaf(c, ai, -s * br);
            sr[i1] = fmaf(c, br,  s * ai);
            si[i1] = fmaf(c, bi, -s * ar);
        }
        __syncthreads();
    }

    // ---- |amp|^2 with ring-permutation #2 folded into the Z signs ----
    float acc[NQ];
    #pragma unroll
    for (int w = 0; w < NQ; ++w) acc[w] = 0.0f;
    for (int j = tid; j < QDIM; j += 256) {
        const float re = sr[j], im = si[j];
        const float pr = re * re + im * im;
        const unsigned pj = ring_perm((unsigned)j);
        #pragma unroll
        for (int w = 0; w < NQ; ++w)
            acc[w] += ((pj >> (9 - w)) & 1) ? -pr : pr;
    }
    #pragma unroll
    for (int w = 0; w < NQ; ++w) {
        float v = acc[w];
        #pragma unroll
        for (int off = 16; off > 0; off >>= 1)
            v += __shfl_down(v, off, 32);             // wave32
        acc[w] = v;
    }
    const int wave = tid >> 5, lane = tid & 31;
    if (lane == 0) {
        #pragma unroll
        for (int w = 0; w < NQ; ++w) red[wave][w] = acc[w];
    }
    __syncthreads();
    if (tid < NQ) {
        float v = 0.0f;
        #pragma unroll
        for (int k = 0; k < 8; ++k) v += red[k][tid];
        meas[(size_t)t * NQ + tid] = v;
    }
}

// ---------------------------------------------------------------------------
// Kernel 2: fused FFN.  One 256-thread block (8 waves) per 32 token rows.
//   Stage h = relu(meas@w1^T + b1) (32 x 1024) in padded LDS (~132 KB of the
//   320 KB WGP LDS -> 2 workgroups/WGP), then out = h @ w2^T + b2 via
//   V_WMMA_F32_16X16X4_F32.  Each wave owns a 32x32 output macro-tile
//   (2 row-tiles x 2 col-tiles = 4 accumulators): every B fragment fetched
//   from L2 feeds two WMMAs, halving L2 traffic vs a 16-row M-tile.
// ---------------------------------------------------------------------------
__global__ __launch_bounds__(256)
void ffn_wmma_kernel(const float* __restrict__ meas,
                     const float* __restrict__ w1, const float* __restrict__ b1,
                     const float* __restrict__ w2, const float* __restrict__ b2,
                     float* __restrict__ out)
{
    extern __shared__ float smem[];
    float* h_s = smem;                   // ROWS * HPAD floats
    float* m_s = smem + ROWS * HPAD;     // ROWS * NQ  floats

    const int tid = threadIdx.x;
    const int r0  = blockIdx.x * ROWS;   // token-row base

    if (tid < ROWS * NQ)
        m_s[tid] = meas[(size_t)r0 * NQ + tid];
    __syncthreads();

    // ---- h tile: K=10 dense layer + ReLU ----
    for (int k = tid; k < FFN; k += 256) {
        float wrow[NQ];
        #pragma unroll
        for (int j = 0; j < NQ; ++j) wrow[j] = w1[(size_t)k * NQ + j];
        const float bk = b1[k];
        #pragma unroll
        for (int r = 0; r < ROWS; ++r) {
            float a = bk;
            #pragma unroll
            for (int j = 0; j < NQ; ++j) a = fmaf(m_s[r * NQ + j], wrow[j], a);
            h_s[r * HPAD + k] = fmaxf(a, 0.0f);
        }
    }
    __syncthreads();

    // ---- WMMA f32 16x16x4 GEMM: A = h (32 x 1024), B = w2^T tiles ----
    const int wave  = tid >> 5;
    const int lane  = tid & 31;
    const int nIdx  = lane & 15;       // A: row M ; B: col N ; D: col N
    const int kHalf = lane >> 4;       // lanes 16-31 carry K+2,K+3
    const int c0    = wave * 32;       // two 16-col tiles per wave

    const float* aptr0 = h_s + (size_t)nIdx * HPAD + (kHalf << 1);  // rows 0-15
    const float* aptr1 = aptr0 + (size_t)16 * HPAD;                 // rows 16-31
    const float* bptr0 = w2 + (size_t)(c0 + nIdx) * FFN + (kHalf << 1);
    const float* bptr1 = bptr0 + (size_t)16 * FFN;

    v8f acc00 = {}; v8f acc01 = {};   // rows 0-15  x cols {c0, c0+16}
    v8f acc10 = {}; v8f acc11 = {};   // rows 16-31 x cols {c0, c0+16}
    for (int kk = 0; kk < FFN / 4; ++kk) {
        const v2f a0 = *(const v2f*)(aptr0 + kk * 4);
        const v2f a1 = *(const v2f*)(aptr1 + kk * 4);
        const v2f b0 = *(const v2f*)(bptr0 + kk * 4);
        const v2f b1v= *(const v2f*)(bptr1 + kk * 4);
        acc00 = __builtin_amdgcn_wmma_f32_16x16x4_f32(false, a0, false, b0,
                                                      (short)0, acc00, false, false);
        acc01 = __builtin_amdgcn_wmma_f32_16x16x4_f32(false, a0, false, b1v,
                                                      (short)0, acc01, false, false);
        acc10 = __builtin_amdgcn_wmma_f32_16x16x4_f32(false, a1, false, b0,
                                                      (short)0, acc10, false, false);
        acc11 = __builtin_amdgcn_wmma_f32_16x16x4_f32(false, a1, false, b1v,
                                                      (short)0, acc11, false, false);
    }

    // ---- epilogue: D layout (lane half -> M = 8*half + v), add b2 ----
    const float bA = b2[c0 + nIdx];
    const float bB = b2[c0 + 16 + nIdx];
    #pragma unroll
    for (int v = 0; v < 8; ++v) {
        const int M = (kHalf << 3) + v;
        out[(size_t)(r0 + M) * EMBED + c0 + nIdx]           = acc00[v] + bA;
        out[(size_t)(r0 + M) * EMBED + c0 + 16 + nIdx]      = acc01[v] + bB;
        out[(size_t)(r0 + 16 + M) * EMBED + c0 + nIdx]      = acc10[v] + bA;
        out[(size_t)(r0 + 16 + M) * EMBED + c0 + 16 + nIdx] = acc11[v] + bB;
    }
}

// ---------------------------------------------------------------------------
extern "C" void kernel_launch(void* const* d_in, const int* in_sizes, int n_in,
                              void* d_out, int out_size, void* d_ws, size_t ws_size,
                              hipStream_t stream)
{
    const float* x  = (const float*)d_in[0];   // (8,2048,256) f32
    const float* qp = (const float*)d_in[1];   // (2,10)       f32
    const float* w1 = (const float*)d_in[2];   // (1024,10)    f32
    const float* b1 = (const float*)d_in[3];   // (1024,)      f32
    const float* w2 = (const float*)d_in[4];   // (256,1024)   f32
    const float* b2 = (const float*)d_in[5];   // (256,)       f32
    float* out  = (float*)d_out;               // (8,2048,256) f32
    float* meas = (float*)d_ws;                // N x 10 f32 scratch (~640 KB)

    const int Ntok = in_sizes[0] / EMBED;      // 16384

    qsim_kernel<<<Ntok, 256, 0, stream>>>(x, qp, meas);

    const size_t smem_bytes = (size_t)(ROWS * HPAD + ROWS * NQ) * sizeof(float);
    ffn_wmma_kernel<<<Ntok / ROWS, 256, smem_bytes, stream>>>(meas, w1, b1, w2, b2, out);
}